// Agreement_Routing_Down_26010321945025
// MI455X (gfx1250) — compile-verified
//
#include <hip/hip_runtime.h>
#include <hip/hip_bf16.h>

// Capsule agreement routing on MI455X (gfx1250, wave32).
// One wave per (b,o,h,w) site; the site's 16x16 fp32 pred tile is loaded once
// (coalesced -> LDS -> WMMA register layouts) and all routing iterations run
// from registers using v_wmma_f32_16x16x4_f32. Routing state (logits, c, s,
// out) is wave-uniform: broadcasts use v_readlane into SGPR-held scalars
// instead of ds_bpermute. Memory-bound: 256MB read + 16MB write ~= 11.7us
// floor at 23.3 TB/s.

typedef float v2f __attribute__((ext_vector_type(2)));
typedef float v8f __attribute__((ext_vector_type(8)));

static __device__ inline v8f wmma_f32_16x16x4(v2f a, v2f b, v8f c) {
  // 8 args: (neg_a, A, neg_b, B, c_mod, C, reuse_a, reuse_b)
  return __builtin_amdgcn_wmma_f32_16x16x4_f32(false, a, false, b, (short)0, c,
                                               false, false);
}

// Wave-uniform broadcast of lane `l`'s value: v_readlane_b32 -> SGPR.
static __device__ inline float readlane_f(float v, int l) {
  union {
    float f;
    int i;
  } u;
  u.f = v;
  u.i = __builtin_amdgcn_readlane(u.i, l);
  return u.f;
}

#define WAVES_PER_BLK 8
#define ROW_STRIDE 18  // 16 + 2 pad floats: 8B-aligned rows, conflict-light

__global__ __launch_bounds__(256) void caps_routing_kernel(
    const float* __restrict__ pred, const float* __restrict__ blog,
    const int* __restrict__ nitp, float* __restrict__ outp, int nsites,
    int bsites) {
  __shared__ float ldsTile[WAVES_PER_BLK][16 * ROW_STRIDE];

  const int lane = threadIdx.x & 31;
  const int wv   = threadIdx.x >> 5;
  const int site = blockIdx.x * WAVES_PER_BLK + wv;
  if (site >= nsites) return;  // wave-uniform

  float* tile = &ldsTile[wv][0];

  // ---- Stage the 1KB tile: 2x global_load_b128 per lane, store to LDS ----
  {
    const int row = lane >> 1, half = (lane & 1) * 8;
    const float4* g =
        reinterpret_cast<const float4*>(pred + (size_t)site * 256 + row * 16 + half);
    float4 x0 = g[0];
    float4 x1 = g[1];
    float* l = tile + row * ROW_STRIDE + half;  // byte offs 72*row{+32}: 8B aligned
    reinterpret_cast<float2*>(l)[0] = make_float2(x0.x, x0.y);
    reinterpret_cast<float2*>(l)[1] = make_float2(x0.z, x0.w);
    reinterpret_cast<float2*>(l)[2] = make_float2(x1.x, x1.y);
    reinterpret_cast<float2*>(l)[3] = make_float2(x1.z, x1.w);
  }
  // Same-wave DS ops are in-order (DScnt); private region -> no barrier.

  const bool lo = (lane & 16) == 0;  // low 16-lane half?
  const int  m  = lane & 15;

  // ---- pred in both WMMA layouts, resident for the whole routing chain ----
  // A chunks: Achunk_k[m,r] = pred[m, 4k+r]
  //   lane<16: (pred[m,4k], pred[m,4k+1]); lane>=16: (pred[m,4k+2], pred[m,4k+3])
  // B chunks: Bchunk_k[r,n] = pred[4k+r, n]
  //   lane<16: (pred[4k,  m], pred[4k+1,m]); lane>=16: (pred[4k+2,m], pred[4k+3,m])
  v2f predA[4], predB[4];
#pragma unroll
  for (int k = 0; k < 4; ++k) {
    const int acol = 4 * k + (lo ? 0 : 2);
    predA[k] = *reinterpret_cast<v2f*>(tile + m * ROW_STRIDE + acol);  // ds b64
    const int brow = 4 * k + (lo ? 0 : 2);
    predB[k].x = tile[brow * ROW_STRIDE + m];
    predB[k].y = tile[(brow + 1) * ROW_STRIDE + m];
  }

  // ---- routing logits (b is [1,O,H,W,I]: shared across batch; uniform) ----
  float bl[16];
  {
    const float* bsrc = blog + (size_t)(site % bsites) * 16;
#pragma unroll
    for (int i = 0; i < 16; ++i) bl[i] = bsrc[i];
  }

  float c[16];     // UNNORMALIZED softmax weights (uniform); 1/sum folded later
  float csum;      // softmax denominator (uniform)
  float outv[16];  // squashed output vector (uniform)
  float laneOut = 0.f;  // lane l holds out[l%16] (for the final store)

  // Unnormalized softmax: c[i] = exp(bl[i]-max), csum = sum. The 1/csum factor
  // is linear through s = sum_i c_i*pred_i, so it is folded into the squash
  // scale (one v_rcp_f32 instead of a division + 16 multiplies).
  auto softmax16 = [&]() {
    float mx = bl[0];
#pragma unroll
    for (int i = 1; i < 16; ++i) mx = fmaxf(mx, bl[i]);
    float sum = 0.f;
#pragma unroll
    for (int i = 0; i < 16; ++i) {
      c[i] = __expf(bl[i] - mx);
      sum += c[i];
    }
    csum = sum;
  };

  // s' = c^T * pred via 4 WMMAs (A = c broadcast over rows, B = pred chunks).
  // D[m,n] = s'[n] for all m -> lane holds s'[lane%16] in acc[0]; broadcast
  // via v_readlane (uniform), then squash with the 1/csum correction.
  auto routeS = [&]() {
    v8f acc = {};
#pragma unroll
    for (int k = 0; k < 4; ++k) {
      v2f ac;
      ac.x = lo ? c[4 * k] : c[4 * k + 2];
      ac.y = lo ? c[4 * k + 1] : c[4 * k + 3];
      acc = wmma_f32_16x16x4(ac, predB[k], acc);
    }
    const float sval = acc[0];
    float sp[16], sqp = 0.f;
#pragma unroll
    for (int d = 0; d < 16; ++d) {
      sp[d] = readlane_f(sval, d);  // uniform s'[d]
      sqp += sp[d] * sp[d];
    }
    const float inv = __frcp_rn(csum);     // s = s' * inv
    const float sq  = sqp * inv * inv;     // |s|^2
    const float g   = (sq / (1.0f + sq)) * __frsqrt_rn(sq + 1e-7f) * inv;
#pragma unroll
    for (int d = 0; d < 16; ++d) outv[d] = g * sp[d];
    laneOut = g * sval;
  };

  // a = pred * out via 4 WMMAs (A = pred chunks, B = out broadcast per row);
  // D[m,n] = a[m]: rows 0..7 live in lanes 0..15, rows 8..15 in lanes 16..31.
  // a is uniform per site -> extract with v_readlane, accumulate into bl.
  auto agree = [&]() {
    v8f acc = {};
#pragma unroll
    for (int k = 0; k < 4; ++k) {
      v2f bo;
      bo.x = lo ? outv[4 * k] : outv[4 * k + 2];
      bo.y = lo ? outv[4 * k + 1] : outv[4 * k + 3];
      acc = wmma_f32_16x16x4(predA[k], bo, acc);
    }
#pragma unroll
    for (int j = 0; j < 8; ++j) {
      bl[j]     += readlane_f(acc[j], 0);   // a[j]
      bl[j + 8] += readlane_f(acc[j], 16);  // a[j+8]
    }
  };

  // initial out from softmax(b), then dynamic routing iterations
  softmax16();
  routeS();
  const int iters = nitp[0];
  for (int t = 0; t < iters; ++t) {
    agree();
    softmax16();
    routeS();
  }

  if (lane < 16) outp[(size_t)site * 16 + lane] = laneOut;
}

extern "C" void kernel_launch(void* const* d_in, const int* in_sizes, int n_in,
                              void* d_out, int out_size, void* d_ws,
                              size_t ws_size, hipStream_t stream) {
  const float* pred = (const float*)d_in[0];  // [B,O,H,W,I,D] f32
  const float* blog = (const float*)d_in[1];  // [1,O,H,W,I]  f32
  const int* nit    = (const int*)d_in[2];    // scalar int
  float* out        = (float*)d_out;          // [B,O,H,W,D]  f32

  const int nsites = out_size / 16;      // B*O*H*W
  const int bsites = in_sizes[1] / 16;   // O*H*W
  const int blocks = (nsites + WAVES_PER_BLK - 1) / WAVES_PER_BLK;

  caps_routing_kernel<<<blocks, 256, 0, stream>>>(pred, blog, nit, out, nsites,
                                                  bsites);
}